// RGBRefLoss_14353780703909
// MI455X (gfx1250) — compile-verified
//
#include <hip/hip_runtime.h>
#include <hip/hip_bf16.h>

// Problem constants (match reference)
static constexpr int kSB   = 8;
static constexpr int kB    = 8192;
static constexpr int kNR   = 8;
static constexpr long long kNPTS = 500000;
static constexpr int kBlk  = 256;                 // 8 waves (wave32)
static constexpr int kBlkX = kB / kBlk;           // 32 blocks per sb

typedef float v2f __attribute__((ext_vector_type(2)));
typedef float v8f __attribute__((ext_vector_type(8)));

// Full-wave (32-lane) sum using V_WMMA_F32_16X16X4_F32.
// A: lane L contributes {v[L], 0} to row (L%16); with B == all-ones,
// C[m][n] = v[m] + v[m+16] for every n, independent of the exact K-slot
// assignment (zeros/ones make all K permutations equivalent).
// Lanes 0-15 hold rows M=0..7 in c[0..7]; lanes 16-31 hold rows M=8..15.
// Sum the 8 accumulators per lane, then fold halves with one shfl_xor.
// Requires EXEC == all ones (callers have no divergence before this point).
__device__ __forceinline__ float wave_sum_f32_wmma(float v) {
  v2f a; a[0] = v;    a[1] = 0.0f;
  v2f b; b[0] = 1.0f; b[1] = 1.0f;
  v8f c = {};
  c = __builtin_amdgcn_wmma_f32_16x16x4_f32(
      /*neg_a=*/false, a, /*neg_b=*/false, b,
      /*c_mod=*/(short)0, c, /*reuse_a=*/false, /*reuse_b=*/false);
  float s = c[0] + c[1] + c[2] + c[3] + c[4] + c[5] + c[6] + c[7];
  s += __shfl_xor(s, 16, 32);
  return s;
}

// Phase 1: per-(sb, 256-b-chunk) partial sums of  num = sum m*|ref-gt|  and
// msum = sum m  (mask is replicated across the 3 channels, so read channel 0).
__global__ __launch_bounds__(kBlk) void rgbref_partial(
    const float* __restrict__ rgb_ref,        // [SB, B, NR, 3]
    const long long* __restrict__ idx_pcloud, // [SB, B] int64, -1 = padded
    const float* __restrict__ color_pcloud,   // [SB, NR, NPTS, 3]
    const float* __restrict__ mask_pcloud,    // [SB, NR, NPTS, 3] (replicated)
    float* __restrict__ ws_num,               // [SB * kBlkX]
    float* __restrict__ ws_den) {             // [SB * kBlkX]
  const int sb = blockIdx.y;
  const int b  = blockIdx.x * kBlk + threadIdx.x;   // always < kB

  const long long idx = idx_pcloud[(size_t)sb * kB + b];
  const float valid   = (idx >= 0) ? 1.0f : 0.0f;
  const long long j   = (idx >= 0) ? idx : 0;       // safe gather index

  // 96 contiguous, 16B-aligned bytes of rgb_ref per thread -> 6 x b128 loads
  const float4* ref4 =
      reinterpret_cast<const float4*>(rgb_ref) + ((size_t)sb * kB + b) * 6;
  float refv[24];
#pragma unroll
  for (int k = 0; k < 6; ++k) {
    float4 t = ref4[k];
    refv[4 * k + 0] = t.x; refv[4 * k + 1] = t.y;
    refv[4 * k + 2] = t.z; refv[4 * k + 3] = t.w;
  }

  float num = 0.0f, msum = 0.0f;
#pragma unroll
  for (int r = 0; r < kNR; ++r) {
    const size_t base = (((size_t)sb * kNR + r) * kNPTS + (size_t)j) * 3;
    const float c0 = color_pcloud[base + 0];
    const float c1 = color_pcloud[base + 1];
    const float c2 = color_pcloud[base + 2];
    const float m  = mask_pcloud[base];             // channel 0 (replicated)
    const float d  = fabsf(refv[3 * r + 0] - c0)
                   + fabsf(refv[3 * r + 1] - c1)
                   + fabsf(refv[3 * r + 2] - c2);
    num  = fmaf(m, d, num);
    msum += m;
  }
  num  *= valid;
  msum *= valid;

  // Wave-level reduction via WMMA (all lanes active here).
  const float nsum = wave_sum_f32_wmma(num);
  const float dsum = wave_sum_f32_wmma(msum);

  __shared__ float sn[kBlk / 32];
  __shared__ float sd[kBlk / 32];
  const int wave = threadIdx.x >> 5;
  if ((threadIdx.x & 31) == 0) { sn[wave] = nsum; sd[wave] = dsum; }
  __syncthreads();
  if (threadIdx.x == 0) {
    float n = 0.0f, d = 0.0f;
#pragma unroll
    for (int i = 0; i < kBlk / 32; ++i) { n += sn[i]; d += sd[i]; }
    ws_num[sb * kBlkX + blockIdx.x] = n;   // fully overwritten every call
    ws_den[sb * kBlkX + blockIdx.x] = d;
  }
}

// Phase 2: one block; wave w == sb reduces its 32 block-partials (WMMA again),
// applies the normalization, thread 0 writes the mean over SB.
__global__ __launch_bounds__(kBlk) void rgbref_final(
    const float* __restrict__ ws_num,
    const float* __restrict__ ws_den,
    float* __restrict__ out) {
  const int wave = threadIdx.x >> 5;   // == sb (8 waves, 8 sb)
  const int lane = threadIdx.x & 31;

  float n = ws_num[wave * kBlkX + lane];
  float d = ws_den[wave * kBlkX + lane];
  n = wave_sum_f32_wmma(n);
  d = wave_sum_f32_wmma(d);

  __shared__ float sl[kSB];
  if (lane == 0) {
    // reference: mask_sum = 3*msum ;  loss_i = mask_sum>0 ?
    //            num / max(mask_sum*3, 1) : mask_sum(==0)
    const float mask_sum = 3.0f * d;
    const float li =
        (mask_sum > 0.0f) ? (n / fmaxf(mask_sum * 3.0f, 1.0f)) : mask_sum;
    sl[wave] = li;
  }
  __syncthreads();
  if (threadIdx.x == 0) {
    float s = 0.0f;
#pragma unroll
    for (int i = 0; i < kSB; ++i) s += sl[i];
    out[0] = s * (1.0f / (float)kSB);
  }
}

extern "C" void kernel_launch(void* const* d_in, const int* in_sizes, int n_in,
                              void* d_out, int out_size, void* d_ws,
                              size_t ws_size, hipStream_t stream) {
  (void)in_sizes; (void)n_in; (void)out_size; (void)ws_size;
  const float*     rgb_ref      = (const float*)d_in[0];
  const long long* idx_pcloud   = (const long long*)d_in[1]; // int64 per ref
  const float*     color_pcloud = (const float*)d_in[2];
  const float*     mask_pcloud  = (const float*)d_in[3];

  float* ws_num = (float*)d_ws;                 // SB*kBlkX floats
  float* ws_den = ws_num + kSB * kBlkX;         // SB*kBlkX floats (2 KB total)

  dim3 grid(kBlkX, kSB);
  rgbref_partial<<<grid, kBlk, 0, stream>>>(rgb_ref, idx_pcloud, color_pcloud,
                                            mask_pcloud, ws_num, ws_den);
  rgbref_final<<<1, kBlk, 0, stream>>>(ws_num, ws_den, (float*)d_out);
}